// TAN_69028714381819
// MI455X (gfx1250) — compile-verified
//
#include <hip/hip_runtime.h>
#include <math.h>

// Problem constants (from reference)
#define B_   8
#define XS   32     // X = Y = Z = 32
#define DD   128    // Dx = Dy = Dz
#define KK   256    // hidden K
#define HH   2      // heads
#define EPSbn 1e-5f

typedef float v2f __attribute__((ext_vector_type(2)));
typedef float v8f __attribute__((ext_vector_type(8)));
typedef unsigned int u32x4 __attribute__((ext_vector_type(4)));
typedef int i32x8 __attribute__((ext_vector_type(8)));
typedef int i32x4 __attribute__((ext_vector_type(4)));

// V_WMMA_F32_16X16X4_F32: D(16x16,f32) = A(16x4,f32) * B(4x16,f32) + C
// Per ISA 7.12.2: A lanes 0-15 hold M=lane, K={0,1}; lanes 16-31 hold M=lane-16, K={2,3}.
// We mirror that K split for B (lane group selects K pair, lo = column N).
// C/D: VGPR j -> M = j + 8*(lane/16), N = lane%16.
static __device__ __forceinline__ v8f wmma4(v2f a, v2f b, v8f c) {
  return __builtin_amdgcn_wmma_f32_16x16x4_f32(
      /*neg_a=*/false, a, /*neg_b=*/false, b,
      /*c_mod=*/(short)0, c, /*reuse_a=*/false, /*reuse_b=*/false);
}

// Low 32 bits of a generic pointer to LDS == LDS byte offset (flat-LDS rule).
static __device__ __forceinline__ unsigned int lds_off(const void* p) {
  return (unsigned int)(uintptr_t)p;
}

// Async memory->LDS copy, 16 bytes per lane. Tracked with ASYNCcnt.
static __device__ __forceinline__ void async_ld_b128(unsigned int lds, const float* g) {
  asm volatile("global_load_async_to_lds_b128 %0, %1, off"
               :: "v"(lds), "v"(g) : "memory");
}
static __device__ __forceinline__ void wait_async() {
  asm volatile("s_wait_asynccnt 0x0" ::: "memory");
}

// TDM: 1-D contiguous dword copy memory->LDS via tensor descriptor (D#).
// group0: count=1 | lds_addr | global_addr[56:0] | type=2
// group1: data_size=4B, tensor_dim0 = tile_dim0 = n dwords, stride = n
static __device__ __forceinline__ void tdm_copy_1d(unsigned int lds_addr,
                                                   const void* gsrc,
                                                   unsigned int n_dw) {
  unsigned long long ga = (unsigned long long)(uintptr_t)gsrc;
  u32x4 g0;
  g0[0] = 1u;                                             // count=1 (valid), user mode
  g0[1] = lds_addr;                                       // lds byte address
  g0[2] = (unsigned int)ga;                                // global_addr[31:0]
  g0[3] = (unsigned int)((ga >> 32) & 0x1FFFFFFu) | (2u << 30); // addr[56:32], type=2
  i32x8 g1;
  g1[0] = (int)(2u << 16);          // data_size = 2 (4 bytes); wg_mask = 0
  g1[1] = (int)(n_dw << 16);        // tensor_dim0[15:0]  -> bits 63:48
  g1[2] = (int)(n_dw >> 16);        // tensor_dim0[31:16] -> bits 79:64
  g1[3] = (int)(n_dw << 16);        // tile_dim0          -> bits 127:112
  g1[4] = 0;                        // tile_dim1 = tile_dim2 = 0 (unused -> 1-D)
  g1[5] = (int)n_dw;                // tensor_dim0_stride[31:0]
  g1[6] = 0;
  g1[7] = 0;
  i32x4 zz4 = {0, 0, 0, 0};
  i32x8 zz8 = {0, 0, 0, 0, 0, 0, 0, 0};
  __builtin_amdgcn_tensor_load_to_lds(g0, g1, zz4, zz4, zz8, 0);
}

// ---------------------------------------------------------------------------
// 1) weight-norm scales: scales[i] = g_i / ||V_i||_F
// ---------------------------------------------------------------------------
__global__ void k_scales(const float* __restrict__ Vx, const float* __restrict__ gx,
                         const float* __restrict__ Vy, const float* __restrict__ gy,
                         const float* __restrict__ Vz, const float* __restrict__ gz,
                         const float* __restrict__ Vh, const float* __restrict__ gh,
                         float* __restrict__ scales) {
  __shared__ float red[256];
  const float* V; const float* g; int n;
  switch (blockIdx.x) {
    case 0:  V = Vx; g = gx; n = KK * DD; break;
    case 1:  V = Vy; g = gy; n = KK * DD; break;
    case 2:  V = Vz; g = gz; n = KK * DD; break;
    default: V = Vh; g = gh; n = HH * KK; break;
  }
  float s = 0.f;
  for (int i = threadIdx.x; i < n; i += 256) { float v = V[i]; s += v * v; }
  red[threadIdx.x] = s; __syncthreads();
  for (int off = 128; off > 0; off >>= 1) {
    if (threadIdx.x < off) red[threadIdx.x] += red[threadIdx.x + off];
    __syncthreads();
  }
  if (threadIdx.x == 0) scales[blockIdx.x] = g[0] / sqrtf(red[0]);
}

// ---------------------------------------------------------------------------
// 2) activation GEMMs: U[m,k] = relu( sum_d In[m,d] * V[k,d] * s + bias[k] )
//    M = B*32 = 256 rows, N = K = 256 cols, inner = 128.
//    grid (M/16, N/128, 3), block 256 (8 waves; wave w -> 16-col slice)
//    A tile staged into LDS by the Tensor Data Mover (wave 0 issues the D#).
// ---------------------------------------------------------------------------
__global__ void k_act_gemm(const float* __restrict__ inX, const float* __restrict__ inY,
                           const float* __restrict__ inZ,
                           const float* __restrict__ Vx, const float* __restrict__ Vy,
                           const float* __restrict__ Vz,
                           const float* __restrict__ bx, const float* __restrict__ by,
                           const float* __restrict__ bz,
                           const float* __restrict__ scales,
                           float* __restrict__ Ux, float* __restrict__ Uy,
                           float* __restrict__ Uz) {
  const float* In; const float* V; const float* bias; float s; float* U;
  if (blockIdx.z == 0)      { In = inX; V = Vx; bias = bx; s = scales[0]; U = Ux; }
  else if (blockIdx.z == 1) { In = inY; V = Vy; bias = by; s = scales[1]; U = Uy; }
  else                      { In = inZ; V = Vz; bias = bz; s = scales[2]; U = Uz; }

  __shared__ __attribute__((aligned(16))) float Alds[16 * DD]; // 16 x 128 (8 KB)
  const int M0 = blockIdx.x * 16;
  const int N0 = blockIdx.y * 128 + (threadIdx.x >> 5) * 16;

  if (threadIdx.x < 32) {               // wave 0 drives the TDM DMA
    tdm_copy_1d(lds_off(Alds), In + (size_t)M0 * DD, 16 * DD);
    __builtin_amdgcn_s_wait_tensorcnt(0);
  }
  __syncthreads();

  const int lane = threadIdx.x & 31, half = lane >> 4, lo = lane & 15;
  const float* vrow = V + (size_t)(N0 + lo) * DD;    // B[k_d][n] = V[n][k_d] * s
  __builtin_prefetch(vrow, 0, 3);                    // global_prefetch_b8

  v8f c = {};
  for (int d0 = 0; d0 < DD; d0 += 4) {
    const int kd = d0 + 2 * half;
    v2f a; a.x = Alds[lo * DD + kd];  a.y = Alds[lo * DD + kd + 1];
    v2f b; b.x = vrow[kd] * s;        b.y = vrow[kd + 1] * s;
    c = wmma4(a, b, c);
  }
#pragma unroll
  for (int j = 0; j < 8; j++) {
    const int m = M0 + j + 8 * half, n = N0 + lo;
    const float v = c[j] + bias[n];
    U[(size_t)m * KK + n] = v > 0.f ? v : 0.f;
  }
}

// ---------------------------------------------------------------------------
// 3) attention scores: for fixed (b,h,x):
//    scr[y,z] = sum_k (sh*Vh[h,k]*Ux[b,x,k]) * Uy[b,y,k] * Uz[b,z,k] + bh[h]
//    32x32 tile per block = 4 waves x (16x16), inner K = 256 -> 64 WMMA/wave
//    grid (32, 2, 8), block 128. Y panel staged via async global->LDS b128.
// ---------------------------------------------------------------------------
__global__ void k_scores(const float* __restrict__ Ux, const float* __restrict__ Uy,
                         const float* __restrict__ Uz,
                         const float* __restrict__ Vh, const float* __restrict__ bh,
                         const float* __restrict__ scales,
                         float* __restrict__ scores) {
  const int x = blockIdx.x, h = blockIdx.y, b = blockIdx.z;
  __shared__ __attribute__((aligned(16))) float wxk[KK];       // Wh-scaled _x row
  __shared__ __attribute__((aligned(16))) float Ylds[XS * KK]; // 32x256 panel (32 KB)
  const float sh = scales[3];

  // async DMA of the 32x256 _y panel into LDS (2048 x 16B chunks, 128 threads)
  const float* ysrc = Uy + (size_t)b * XS * KK;
  const unsigned int ybase = lds_off(Ylds);
  for (int i = threadIdx.x; i < (XS * KK) / 4; i += 128)
    async_ld_b128(ybase + (unsigned int)i * 16u, ysrc + (size_t)i * 4);

  for (int k = threadIdx.x; k < KK; k += 128)
    wxk[k] = Vh[h * KK + k] * sh * Ux[((size_t)b * XS + x) * KK + k];

  wait_async();
  __syncthreads();

  const int wave = threadIdx.x >> 5;
  const int wy16 = (wave >> 1) * 16, wz16 = (wave & 1) * 16;
  const int lane = threadIdx.x & 31, half = lane >> 4, lo = lane & 15;
  const float* yrow = &Ylds[(wy16 + lo) * KK];
  const float* zrow = Uz + ((size_t)b * XS + wz16 + lo) * KK;

  v8f c = {};
  for (int k0 = 0; k0 < KK; k0 += 4) {
    const int kd = k0 + 2 * half;
    v2f a; a.x = wxk[kd] * yrow[kd];      a.y = wxk[kd + 1] * yrow[kd + 1];
    v2f bb; bb.x = zrow[kd];              bb.y = zrow[kd + 1];
    c = wmma4(a, bb, c);
  }
  const float bhv = bh[h];
  const size_t base = (((size_t)b * HH + h) * XS + x) * 1024;
#pragma unroll
  for (int j = 0; j < 8; j++) {
    const int yy = wy16 + j + 8 * half, zz = wz16 + lo;
    scores[base + yy * 32 + zz] = c[j] + bhv;
  }
}

// ---------------------------------------------------------------------------
// 4) softmax over 32768 elems per (b,h); writes post-softmax att into d_out
// ---------------------------------------------------------------------------
__global__ void k_softmax(const float* __restrict__ scores, float* __restrict__ att_out) {
  __shared__ float red[1024];
  const int bh = blockIdx.x, t = threadIdx.x;
  const float* src = scores + (size_t)bh * 32768;
  float* dst = att_out + (size_t)bh * 32768;

  float m = -3.4e38f;
  for (int i = t; i < 32768; i += 1024) m = fmaxf(m, src[i]);
  red[t] = m; __syncthreads();
  for (int off = 512; off > 0; off >>= 1) {
    if (t < off) red[t] = fmaxf(red[t], red[t + off]);
    __syncthreads();
  }
  m = red[0]; __syncthreads();

  float s = 0.f;
  for (int i = t; i < 32768; i += 1024) s += __expf(src[i] - m);
  red[t] = s; __syncthreads();
  for (int off = 512; off > 0; off >>= 1) {
    if (t < off) red[t] += red[t + off];
    __syncthreads();
  }
  const float inv = 1.f / red[0];
  for (int i = t; i < 32768; i += 1024) dst[i] = __expf(src[i] - m) * inv;
}

// ---------------------------------------------------------------------------
// 5) head-sum: attsum[b,i] = att[b,0,i] + att[b,1,i]
// ---------------------------------------------------------------------------
__global__ void k_attsum(const float* __restrict__ att, float* __restrict__ attsum) {
  const size_t gid = (size_t)blockIdx.x * 256 + threadIdx.x;   // B*32768 total
  const size_t b = gid >> 15, i = gid & 32767;
  attsum[gid] = att[(b * 2 + 0) * 32768 + i] + att[(b * 2 + 1) * 32768 + i];
}

// ---------------------------------------------------------------------------
// 6) R[b,xy,k] = sum_z attsum[b,xy,z] * Uz[b,z,k]   (per-b 1024x32 @ 32x256)
//    grid (64, 2, 8), block 256. A tile staged via async global->LDS b128.
// ---------------------------------------------------------------------------
__global__ void k_rgemm(const float* __restrict__ attsum, const float* __restrict__ Uz,
                        float* __restrict__ R) {
  const int b = blockIdx.z;
  const int M0 = blockIdx.x * 16;
  const int N0 = blockIdx.y * 128 + (threadIdx.x >> 5) * 16;
  __shared__ __attribute__((aligned(16))) float Alds[16 * 32]; // 2 KB
  const float* asrc = attsum + (size_t)b * 32768 + (size_t)M0 * 32;
  const unsigned int abase = lds_off(Alds);
  if (threadIdx.x < 128)      // waves 0-3: 128 x 16B async chunks
    async_ld_b128(abase + threadIdx.x * 16u, asrc + (size_t)threadIdx.x * 4);
  wait_async();
  __syncthreads();

  const int lane = threadIdx.x & 31, half = lane >> 4, lo = lane & 15;
  v8f c = {};
  for (int z0 = 0; z0 < 32; z0 += 4) {
    const int kd = z0 + 2 * half;
    v2f a; a.x = Alds[lo * 32 + kd]; a.y = Alds[lo * 32 + kd + 1];
    v2f bb;
    bb.x = Uz[((size_t)b * 32 + kd) * KK + N0 + lo];
    bb.y = Uz[((size_t)b * 32 + kd + 1) * KK + N0 + lo];
    c = wmma4(a, bb, c);
  }
#pragma unroll
  for (int j = 0; j < 8; j++) {
    const int m = M0 + j + 8 * half;
    R[((size_t)b * 1024 + m) * KK + N0 + lo] = c[j];
  }
}

// ---------------------------------------------------------------------------
// 7) logits[b,k] = sum_{x,y} Ux[b,x,k]*Uy[b,y,k]*R[b,x*32+y,k]
// ---------------------------------------------------------------------------
__global__ void k_logits(const float* __restrict__ Ux, const float* __restrict__ Uy,
                         const float* __restrict__ R, float* __restrict__ logits) {
  const int b = blockIdx.x, k = threadIdx.x;
  float acc = 0.f;
  for (int x = 0; x < 32; x++) {
    const float ux = Ux[((size_t)b * 32 + x) * KK + k];
    const float* Rrow = R + ((size_t)b * 1024 + (size_t)x * 32) * KK + k;
    for (int y = 0; y < 32; y++)
      acc += ux * Uy[((size_t)b * 32 + y) * KK + k] * Rrow[(size_t)y * KK];
  }
  logits[b * KK + k] = acc;
}

// ---------------------------------------------------------------------------
// 8) BatchNorm1d over B=8 (training-mode, biased var) -> d_out[0:2048]
// ---------------------------------------------------------------------------
__global__ void k_bn(const float* __restrict__ logits, const float* __restrict__ gamma,
                     const float* __restrict__ beta, float* __restrict__ out) {
  const int k = threadIdx.x;
  float v[B_]; float mean = 0.f;
#pragma unroll
  for (int b = 0; b < B_; b++) { v[b] = logits[b * KK + k]; mean += v[b]; }
  mean *= (1.f / B_);
  float var = 0.f;
#pragma unroll
  for (int b = 0; b < B_; b++) { const float d = v[b] - mean; var += d * d; }
  var *= (1.f / B_);
  const float inv = rsqrtf(var + EPSbn);
  const float g = gamma[k], be = beta[k];
#pragma unroll
  for (int b = 0; b < B_; b++) out[b * KK + k] = (v[b] - mean) * inv * g + be;
}

// ---------------------------------------------------------------------------
extern "C" void kernel_launch(void* const* d_in, const int* in_sizes, int n_in,
                              void* d_out, int out_size, void* d_ws, size_t ws_size,
                              hipStream_t stream) {
  const float* x    = (const float*)d_in[0];
  const float* y    = (const float*)d_in[1];
  const float* z    = (const float*)d_in[2];
  const float* Vx   = (const float*)d_in[3];
  const float* gx   = (const float*)d_in[4];
  const float* bx   = (const float*)d_in[5];
  const float* Vy   = (const float*)d_in[6];
  const float* gy   = (const float*)d_in[7];
  const float* by   = (const float*)d_in[8];
  const float* Vz   = (const float*)d_in[9];
  const float* gz   = (const float*)d_in[10];
  const float* bz   = (const float*)d_in[11];
  const float* Vh   = (const float*)d_in[12];
  const float* gh   = (const float*)d_in[13];
  const float* bh   = (const float*)d_in[14];
  const float* gam  = (const float*)d_in[15];
  const float* bet  = (const float*)d_in[16];

  float* out = (float*)d_out;                 // [0,2048) = bn out
  float* att = out + (size_t)B_ * KK;         // [2048, 2048+524288) = softmax att

  // workspace layout (floats)
  float* ws      = (float*)d_ws;
  float* scales  = ws;                                   // 4
  float* Ux      = ws + 16;                              // 65536
  float* Uy      = Ux + (size_t)B_ * XS * KK;            // 65536
  float* Uz      = Uy + (size_t)B_ * XS * KK;            // 65536
  float* scores  = Uz + (size_t)B_ * XS * KK;            // 524288
  float* attsum  = scores + (size_t)B_ * HH * 32768;     // 262144
  float* Rbuf    = attsum + (size_t)B_ * 32768;          // 2097152
  float* logits  = Rbuf + (size_t)B_ * 1024 * KK;        // 2048

  k_scales<<<4, 256, 0, stream>>>(Vx, gx, Vy, gy, Vz, gz, Vh, gh, scales);

  k_act_gemm<<<dim3((B_ * XS) / 16, KK / 128, 3), 256, 0, stream>>>(
      x, y, z, Vx, Vy, Vz, bx, by, bz, scales, Ux, Uy, Uz);

  k_scores<<<dim3(XS, HH, B_), 128, 0, stream>>>(Ux, Uy, Uz, Vh, bh, scales, scores);

  k_softmax<<<B_ * HH, 1024, 0, stream>>>(scores, att);

  k_attsum<<<(B_ * 32768) / 256, 256, 0, stream>>>(att, attsum);

  k_rgemm<<<dim3(1024 / 16, KK / 128, B_), 256, 0, stream>>>(attsum, Uz, Rbuf);

  k_logits<<<B_, KK, 0, stream>>>(Ux, Uy, Rbuf, logits);

  k_bn<<<1, KK, 0, stream>>>(logits, gam, bet, out);
}